// ForwardEulerNeuralSolver_37065567764797
// MI455X (gfx1250) — compile-verified
//
#include <hip/hip_runtime.h>
#include <math.h>

typedef __attribute__((ext_vector_type(2))) float v2f;
typedef __attribute__((ext_vector_type(8))) float v8f;

// Problem constants (match reference)
constexpr int NB   = 2;
constexpr int NN   = 131072;
constexpr int DD   = 24;      // total channels
constexpr int KNB  = 4;       // neighbours (incl self)
constexpr int HID  = 128;
constexpr int DLAT = 16;
constexpr int TPB  = NN / 16; // 16-node tiles per batch = 8192

// ws layout (floats)
constexpr size_t XCOUNT   = (size_t)NB * NN * DD;   // 6291456 floats ping buffer
constexpr size_t W1S_OFF  = XCOUNT;
constexpr size_t W1S_F2   = 24 * 8 * 32;             // kk * j * lane
constexpr size_t W2S_F2   = 32 * 32;                 // kk2 * lane
constexpr size_t WSW_FLTS = 2 * (W1S_F2 + W2S_F2);   // 14336 floats

// LDS layout (floats): [w1s 12288][w2s 2048][per-wave scratch 2048 x 8]
constexpr int LDS_W2_OFF    = 12288;
constexpr int LDS_WAVE_OFF  = 14336;
constexpr int LDS_WAVE_SZ   = 2048;            // 8KB: Z uses 1536, H uses 2048
constexpr int LDS_TOTAL_B   = (LDS_WAVE_OFF + 8 * LDS_WAVE_SZ) * 4;  // 122880 B

__device__ __forceinline__ float gelu_tanh(float x) {
    // jax.nn.gelu default (approximate=True)
    float u = 0.7978845608028654f * (x + 0.044715f * x * x * x);
    return 0.5f * x * (1.0f + tanhf(u));
}

// ---------------------------------------------------------------------------
// Pre-swizzle W1/W2 into per-lane WMMA B-fragment order.
// f32 16x16x4 B layout: lanes 0-15 hold N=0..15 with (VGPR0,VGPR1)=(K,K+1);
// lanes 16-31 hold N=0..15 with (K+2,K+3).  Pack each lane's pair adjacent
// so a fragment is one ds_load_b64.
// ---------------------------------------------------------------------------
__global__ __launch_bounds__(256) void swizzle_weights(
    const float* __restrict__ W1, const float* __restrict__ W2,
    float* __restrict__ wsw)
{
    int t = blockIdx.x * 256 + threadIdx.x;
    if (t < (int)W1S_F2) {
        int kk  = t >> 8;
        int rem = t & 255;
        int j    = rem >> 5;
        int l    = rem & 31;
        int half = l >> 4;
        int n    = l & 15;
        int row0 = 4 * kk + 2 * half;
        int col  = 16 * j + n;
        float2 v = make_float2(W1[row0 * HID + col], W1[(row0 + 1) * HID + col]);
        reinterpret_cast<float2*>(wsw)[t] = v;
    } else if (t < (int)(W1S_F2 + W2S_F2)) {
        int s    = t - (int)W1S_F2;
        int kk2  = s >> 5;
        int l    = s & 31;
        int half = l >> 4;
        int n    = l & 15;
        int row0 = 4 * kk2 + 2 * half;
        float2 v = make_float2(W2[row0 * DLAT + n], W2[(row0 + 1) * DLAT + n]);
        reinterpret_cast<float2*>(wsw)[(int)W1S_F2 + s] = v;
    }
}

// ---------------------------------------------------------------------------
// One forward-Euler step.  One wave = one 16-node tile; 8 waves per WG.
// Fragment fetches are double-buffered so DS latency overlaps the WMMAs.
// ---------------------------------------------------------------------------
__global__ __launch_bounds__(256) void euler_step(
    const float* __restrict__ xin, float* __restrict__ xout,
    const float* __restrict__ wsw,              // swizzled W1|W2 (global, L2-hot)
    const float* __restrict__ b1, const float* __restrict__ b2,
    const float* __restrict__ tf, const int* __restrict__ nbr,
    int step)
{
    extern __shared__ float smem[];
    const int tid  = threadIdx.x;
    const int wave = tid >> 5;
    const int lane = tid & 31;

    // --- cooperative load of swizzled weights into LDS (14336 floats) ---
    {
        const float4* src = reinterpret_cast<const float4*>(wsw);
        float4*       dst = reinterpret_cast<float4*>(smem);
        #pragma unroll
        for (int i = 0; i < (int)(WSW_FLTS / 4) / 256; ++i)
            dst[tid + i * 256] = src[tid + i * 256];
    }
    __syncthreads();

    float* zh = smem + LDS_WAVE_OFF + wave * LDS_WAVE_SZ;

    const int tile      = blockIdx.x * 8 + wave;
    const int batch     = tile / TPB;
    const int node_base = (tile % TPB) * 16;
    const size_t xbatch = (size_t)batch * NN * DD;

    // --- gather Z: 64 rows (m,k) of 24 floats; 2 rows per lane ---
    #pragma unroll
    for (int rr = 0; rr < 2; ++rr) {
        int r = lane + rr * 32;
        int m = r >> 2, k = r & 3;
        int idx = nbr[(node_base + m) * KNB + k];
        const float4* src = reinterpret_cast<const float4*>(xin + xbatch + (size_t)idx * DD);
        float4*       dst = reinterpret_cast<float4*>(zh + m * 96 + k * 24);
        #pragma unroll
        for (int q = 0; q < 6; ++q) dst[q] = src[q];
    }
    __syncthreads();

    const int n0   = lane & 15;
    const int half = lane >> 4;

    const float2* zf2  = reinterpret_cast<const float2*>(zh);
    const float2* w1f2 = reinterpret_cast<const float2*>(smem);
    const float2* w2f2 = reinterpret_cast<const float2*>(smem + LDS_W2_OFF);

    // --- layer 1: (16x96) @ (96x128) ; 24 k-steps x 8 n-tiles ---
    v8f acc[8];
    #pragma unroll
    for (int j = 0; j < 8; ++j) {
        float bv = b1[16 * j + n0];
        acc[j] = (v8f){bv, bv, bv, bv, bv, bv, bv, bv};
    }

    auto loadA1 = [&](int kk) -> v2f {
        float2 t = zf2[n0 * 48 + 2 * kk + half];
        return (v2f){t.x, t.y};
    };
    auto loadB1 = [&](int kk, v2f* b) {
        #pragma unroll
        for (int j = 0; j < 8; ++j) {
            float2 t = w1f2[(kk * 8 + j) * 32 + lane];
            b[j] = (v2f){t.x, t.y};
        }
    };
    auto mma1 = [&](v2f a, const v2f* b) {
        #pragma unroll
        for (int j = 0; j < 8; ++j)
            acc[j] = __builtin_amdgcn_wmma_f32_16x16x4_f32(
                false, a, false, b[j], (short)0, acc[j], false, false);
    };

    {
        v2f aA, aB;
        v2f bA[8], bB[8];
        aA = loadA1(0); loadB1(0, bA);
        #pragma unroll
        for (int kk = 0; kk < 24; kk += 2) {
            aB = loadA1(kk + 1); loadB1(kk + 1, bB);   // prefetch next group
            mma1(aA, bA);
            if (kk + 2 < 24) { aA = loadA1(kk + 2); loadB1(kk + 2, bA); }
            mma1(aB, bB);
        }
    }

    // --- GELU, write H (16x128) into the same per-wave scratch (Z dead) ---
    #pragma unroll
    for (int j = 0; j < 8; ++j) {
        #pragma unroll
        for (int r = 0; r < 8; ++r) {
            float h = gelu_tanh(acc[j][r]);
            int row = half ? (r + 8) : r;
            zh[row * HID + 16 * j + n0] = h;
        }
    }
    __syncthreads();

    // --- layer 2: (16x128) @ (128x16) ; 32 k-steps, two accumulators ---
    float bv2 = b2[n0];
    v8f acc2a = (v8f){bv2, bv2, bv2, bv2, bv2, bv2, bv2, bv2};
    v8f acc2b = (v8f){0.f, 0.f, 0.f, 0.f, 0.f, 0.f, 0.f, 0.f};

    auto loadA2 = [&](int kk2) -> v2f {
        float2 t = zf2[n0 * 64 + 2 * kk2 + half];
        return (v2f){t.x, t.y};
    };
    auto loadB2 = [&](int kk2) -> v2f {
        float2 t = w2f2[kk2 * 32 + lane];
        return (v2f){t.x, t.y};
    };

    {
        v2f aA = loadA2(0), bA = loadB2(0);
        v2f aB, bB;
        #pragma unroll
        for (int kk2 = 0; kk2 < 32; kk2 += 2) {
            aB = loadA2(kk2 + 1); bB = loadB2(kk2 + 1);
            acc2a = __builtin_amdgcn_wmma_f32_16x16x4_f32(
                false, aA, false, bA, (short)0, acc2a, false, false);
            if (kk2 + 2 < 32) { aA = loadA2(kk2 + 2); bA = loadB2(kk2 + 2); }
            acc2b = __builtin_amdgcn_wmma_f32_16x16x4_f32(
                false, aB, false, bB, (short)0, acc2b, false, false);
        }
    }
    v8f acc2 = acc2a + acc2b;

    // --- masked Euler update + ancillary pass-through ---
    float t  = (float)step;                   // STEPSIZE = 1.0
    float ms = tf[batch] - t;
    ms = fminf(fmaxf(ms, 0.0f), 1.0f);

    const size_t obase = xbatch + (size_t)node_base * DD;
    #pragma unroll
    for (int r = 0; r < 8; ++r) {
        int m = half ? (r + 8) : r;
        size_t a = obase + (size_t)m * DD + n0;   // latent channel n0
        xout[a] = xin[a] + ms * acc2[r];
    }
    // ancillary channels 16..23: straight copy (dx padded with zeros)
    {
        int m2 = lane >> 1;
        int c0 = DLAT + (lane & 1) * 4;
        size_t a2 = obase + (size_t)m2 * DD + c0;
        *reinterpret_cast<float4*>(xout + a2) =
            *reinterpret_cast<const float4*>(xin + a2);
    }
}

// ---------------------------------------------------------------------------
extern "C" void kernel_launch(void* const* d_in, const int* in_sizes, int n_in,
                              void* d_out, int out_size, void* d_ws, size_t ws_size,
                              hipStream_t stream) {
    (void)in_sizes; (void)n_in; (void)out_size; (void)ws_size;

    const float* x   = (const float*)d_in[0];
    const float* tf  = (const float*)d_in[1];
    const float* W1  = (const float*)d_in[2];
    const float* b1  = (const float*)d_in[3];
    const float* W2  = (const float*)d_in[4];
    const float* b2  = (const float*)d_in[5];
    const int*   nbr = (const int*)d_in[6];

    float* wsA = (float*)d_ws;            // ping buffer (25.2 MB)
    float* wsw = wsA + W1S_OFF;           // swizzled weights (14336 floats)
    float* out = (float*)d_out;

    // pre-swizzle weights into WMMA fragment order
    swizzle_weights<<<28, 256, 0, stream>>>(W1, W2, wsw);

    const int grid = (NB * TPB) / 8;      // 2048 workgroups, 8 tiles each
    // 4 forward-Euler steps; device-wide sync via separate launches,
    // ping-pong between ws and d_out (final step lands in d_out).
    euler_step<<<grid, 256, LDS_TOTAL_B, stream>>>(x,   wsA, wsw, b1, b2, tf, nbr, 0);
    euler_step<<<grid, 256, LDS_TOTAL_B, stream>>>(wsA, out, wsw, b1, b2, tf, nbr, 1);
    euler_step<<<grid, 256, LDS_TOTAL_B, stream>>>(out, wsA, wsw, b1, b2, tf, nbr, 2);
    euler_step<<<grid, 256, LDS_TOTAL_B, stream>>>(wsA, out, wsw, b1, b2, tf, nbr, 3);
}